// CapsNet_76476187673275
// MI455X (gfx1250) — compile-verified
//
#include <hip/hip_runtime.h>
#include <math.h>

// ---------------- problem constants (from reference) ----------------
#define NB     8        // batch
#define B_     32
#define C_     32
#define KK     3
#define H      4
#define HH     16
#define WIN    14
#define W_     6        // (14-3)/2+1
#define WW     36       // W_*W_
#define BKK    288      // KK*KK*B_
#define CWW    1152     // C_*WW
#define ITER   3
#define EPSF   1e-10f
#define LN_2PI 1.8378770664093453f

typedef float v2f __attribute__((ext_vector_type(2)));
typedef float v8f __attribute__((ext_vector_type(8)));

// ---------------------------------------------------------------
// Kernel 1: activation patch extraction
// ap[b][i=(B,u,v)][s=(X,Y)] = act[b][B][2Y+u][2X+v]
// ---------------------------------------------------------------
__global__ void ap_kernel(const float* __restrict__ act, float* __restrict__ ap)
{
    int idx = blockIdx.x * blockDim.x + threadIdx.x;
    if (idx >= NB * BKK * WW) return;
    int s  = idx % WW;
    int i  = (idx / WW) % BKK;
    int bb = idx / (WW * BKK);
    int Bc = i / 9;
    int uv = i % 9;
    int u = uv / 3, v = uv % 3;
    int X = s / W_, Y = s % W_;
    ap[idx] = act[((bb * B_ + Bc) * WIN + (2 * Y + u)) * WIN + (2 * X + v)];
}

// ---------------------------------------------------------------
// Kernel 2: votes via V_WMMA_F32_16X16X4_F32, LDS-staged coalesced
// writeback.
// Per (B,u,v): D(128x1152) = A(128x4) * Bm(4x1152)
//   A[m=(c,p)][q]      = W[B][u][v][c][p][q]
//   Bm[q][n=(b,X,Y,r)] = poses[b][B][2Y+u][2X+v][q*4+r]
//   D -> V[b][i=(B,u,v)][c*36 + X*6+Y][p*4+r]
// Each N-tile (16 cols) spans exactly one batch bb and 4 consecutive
// spatial positions (36 % 4 == 0), so the 128x16 block is 32 chunks
// of 256 contiguous bytes in V -> global_store_b128, full cachelines.
// Grid: BKK * NSPLIT blocks, 256 threads (8 waves, one M-tile each).
// ---------------------------------------------------------------
#define NSPLIT 8          // 72 N-tiles split across 8 blocks -> 9 tiles/block

__global__ __launch_bounds__(256)
void votes_wmma_kernel(const float* __restrict__ poses,
                       const float* __restrict__ Wt,
                       float* __restrict__ V)
{
    __shared__ float tile[128 * 16];     // one 128x16 f32 D block (8 KB)

    const int blk   = blockIdx.x;
    const int chunk = blk % NSPLIT;
    const int buv   = blk / NSPLIT;
    const int Bc    = buv / 9;
    const int uv    = buv % 9;
    const int u = uv / 3, v = uv % 3;

    const int wave = threadIdx.x >> 5;   // M-tile id (0..7)
    const int lane = threadIdx.x & 31;
    const int l16  = lane & 15;
    const int hi   = lane >> 4;          // 0: K={0,1}, 1: K={2,3}

    // ---- A fragment (16x4 f32, 2 VGPRs/lane per ISA layout) ----
    const int m = wave * 16 + l16;       // global row 0..127
    const int c = m >> 2;
    const int p = m & 3;
    const float* Wb = Wt + ((((size_t)Bc * 3 + u) * 3 + v) * C_ + c) * (H * H) + p * H;
    v2f afrag;
    afrag.x = Wb[2 * hi + 0];
    afrag.y = Wb[2 * hi + 1];

    const int i_bkk = (Bc * 3 + u) * 3 + v;

    for (int k = 0; k < (72 / NSPLIT); ++k) {
        const int nt = chunk * (72 / NSPLIT) + k;
        // ---- B fragment (4x16 f32, 2 VGPRs/lane) ----
        const int n    = nt * 16 + l16;          // global col 0..1151
        const int r    = n & 3;
        const int srem = n >> 2;
        const int s    = srem % WW;              // X*6+Y
        const int bb   = srem / WW;
        const int X = s / W_, Y = s % W_;
        const float* Pp = poses +
            (((size_t)(bb * B_ + Bc) * WIN + (2 * Y + u)) * WIN + (2 * X + v)) * HH;
        v2f bfrag;
        bfrag.x = Pp[(2 * hi + 0) * H + r];
        bfrag.y = Pp[(2 * hi + 1) * H + r];

        v8f acc = {0.f, 0.f, 0.f, 0.f, 0.f, 0.f, 0.f, 0.f};
        acc = __builtin_amdgcn_wmma_f32_16x16x4_f32(
            /*neg_a=*/false, afrag, /*neg_b=*/false, bfrag,
            /*c_mod=*/(short)0, acc, /*reuse_a=*/false, /*reuse_b=*/false);

        // ---- stage D in LDS per ISA C/D layout: VGPR g -> M = g / g+8 ----
        #pragma unroll
        for (int g = 0; g < 8; ++g) {
            const int mg = wave * 16 + (hi ? g + 8 : g);
            tile[mg * 16 + l16] = acc[g];
        }
        __syncthreads();

        // ---- cooperative coalesced writeback: 2048 floats = 512 float4 ----
        const int s0  = (nt * 4) % WW;           // first spatial pos of tile
        const int bb0 = (nt * 4) / WW;           // batch of this tile
        float* Vbase = V + ((size_t)bb0 * BKK + i_bkk) * CWW * HH;
        #pragma unroll
        for (int half = 0; half < 2; ++half) {
            const int f4   = threadIdx.x + half * 256;  // 0..511
            const int cc   = f4 >> 4;                   // output capsule c
            const int rem  = f4 & 15;
            const int soff = rem >> 2;                  // 0..3 spatial offset
            const int hq   = rem & 3;                   // p (hh quad)
            const float4 val = *(const float4*)&tile[(cc * 4 + hq) * 16 + soff * 4];
            *(float4*)&Vbase[((size_t)cc * WW + s0 + soff) * HH + hq * 4] = val;
        }
        __syncthreads();
    }
}

// ---------------------------------------------------------------
// Kernel 3: EM stats (one pass over V per iteration)
// Per (b, j): sumRw, mu[hh], sigma_sq[hh] via moments, then a.
// Grid: NB*CWW blocks, 64 threads.
// ---------------------------------------------------------------
__global__ __launch_bounds__(64)
void em_stats_kernel(const float* __restrict__ V,
                     const float* __restrict__ R,
                     const float* __restrict__ ap,
                     const float* __restrict__ beta_v,
                     const float* __restrict__ beta_a,
                     const float* __restrict__ lambda_p,
                     int first_iter,
                     float* __restrict__ mu_out,   // [NB][CWW][HH]
                     float* __restrict__ sig_out,  // [NB][CWW][HH]
                     float* __restrict__ a_out)    // [NB][CWW]
{
    const int j  = blockIdx.x % CWW;
    const int bb = blockIdx.x / CWW;
    const int c  = j / WW;
    const int s  = j % WW;
    const int t  = threadIdx.x;

    float s0 = 0.f;
    float s1[HH], s2[HH];
    #pragma unroll
    for (int h = 0; h < HH; ++h) { s1[h] = 0.f; s2[h] = 0.f; }

    for (int i = t; i < BKK; i += 64) {
        const float r  = first_iter ? (1.0f / C_)
                                    : R[((size_t)bb * BKK + i) * CWW + j];
        const float rw = r * ap[(bb * BKK + i) * WW + s];
        const float4* vp =
            (const float4*)(V + (((size_t)bb * BKK + i) * CWW + j) * HH);
        #pragma unroll
        for (int q = 0; q < 4; ++q) {
            float4 vv = vp[q];
            s1[q * 4 + 0] += rw * vv.x;  s2[q * 4 + 0] += rw * vv.x * vv.x;
            s1[q * 4 + 1] += rw * vv.y;  s2[q * 4 + 1] += rw * vv.y * vv.y;
            s1[q * 4 + 2] += rw * vv.z;  s2[q * 4 + 2] += rw * vv.z * vv.z;
            s1[q * 4 + 3] += rw * vv.w;  s2[q * 4 + 3] += rw * vv.w * vv.w;
        }
        s0 += rw;
    }

    __shared__ float buf[33 * 64];
    __shared__ float stats[33];
    __shared__ float logbuf[HH];
    buf[0 * 64 + t] = s0;
    #pragma unroll
    for (int h = 0; h < HH; ++h) {
        buf[(1 + h) * 64 + t]  = s1[h];
        buf[(17 + h) * 64 + t] = s2[h];
    }
    __syncthreads();
    if (t < 33) {
        float acc = 0.f;
        for (int n = 0; n < 64; ++n) acc += buf[t * 64 + n];
        stats[t] = acc;
    }
    __syncthreads();

    const float sumR = stats[0];
    if (t < HH) {
        const float mu = stats[1 + t] / sumR;
        float sq = stats[17 + t] / sumR - mu * mu;
        mu_out[((size_t)bb * CWW + j) * HH + t]  = mu;
        sig_out[((size_t)bb * CWW + j) * HH + t] = sq;
        logbuf[t] = logf(sqrtf(sq) + EPSF);
    }
    __syncthreads();
    if (t == 0) {
        float sl = 0.f;
        #pragma unroll
        for (int h = 0; h < HH; ++h) sl += logbuf[h];
        const float cost = sumR * ((float)HH * beta_v[c] + sl);
        const float x    = lambda_p[0] * (beta_a[c] - cost);
        a_out[(size_t)bb * CWW + j] = 1.f / (1.f + expf(-x));
    }
}

// ---------------------------------------------------------------
// Kernel 4: R update (iterations 0..ITER-2)
// Per (b, i): ap_j = a[j] * sum_hh exp(ln_p); R = ap_j/(sum_j ap + eps) + eps
// Grid: NB*BKK blocks, 256 threads. LDS holds the 1152 ap_j values.
// ---------------------------------------------------------------
__global__ __launch_bounds__(256)
void em_rupdate_kernel(const float* __restrict__ V,
                       const float* __restrict__ mu,
                       const float* __restrict__ sig,
                       const float* __restrict__ a,
                       float* __restrict__ R)
{
    const int i  = blockIdx.x % BKK;
    const int bb = blockIdx.x / BKK;
    const int t  = threadIdx.x;

    __shared__ float apbuf[CWW];
    __shared__ float red[8];

    float part = 0.f;
    for (int j = t; j < CWW; j += 256) {
        const float* vp = V + (((size_t)bb * BKK + i) * CWW + j) * HH;
        const float* mp = mu + ((size_t)bb * CWW + j) * HH;
        const float* sp = sig + ((size_t)bb * CWW + j) * HH;
        __builtin_prefetch(vp + 256 * HH, 0, 1);   // global_prefetch_b8
        float psum = 0.f;
        #pragma unroll
        for (int h = 0; h < HH; ++h) {
            const float sq   = sp[h];
            const float logs = logf(sqrtf(sq) + EPSF);
            const float d    = vp[h] - mp[h];
            const float lnp  = -d * d / (2.f * sq) - logs - 0.5f * LN_2PI;
            psum += expf(lnp);
        }
        const float apv = a[(size_t)bb * CWW + j] * psum;
        apbuf[j] = apv;
        part += apv;
    }
    // block reduction (wave32 shuffles + LDS across 8 waves)
    for (int off = 16; off > 0; off >>= 1) part += __shfl_down(part, off, 32);
    if ((t & 31) == 0) red[t >> 5] = part;
    __syncthreads();
    float tot = 0.f;
    #pragma unroll
    for (int wv = 0; wv < 8; ++wv) tot += red[wv];
    const float inv = 1.f / (tot + EPSF);

    for (int j = t; j < CWW; j += 256) {
        R[((size_t)bb * BKK + i) * CWW + j] = apbuf[j] * inv + EPSF;
    }
}

// ---------------------------------------------------------------
// Host launcher
// Workspace layout (floats):
//   V     : NB*BKK*CWW*HH = 42,467,328   (~170 MB -> resident in 192MB L2)
//   R     : NB*BKK*CWW    =  2,654,208
//   ap    : NB*BKK*WW     =     82,944
//   mu    : NB*CWW*HH     =    147,456
//   sigma : NB*CWW*HH     =    147,456
//   a     : NB*CWW        =      9,216
// ---------------------------------------------------------------
extern "C" void kernel_launch(void* const* d_in, const int* in_sizes, int n_in,
                              void* d_out, int out_size, void* d_ws, size_t ws_size,
                              hipStream_t stream)
{
    (void)in_sizes; (void)n_in; (void)out_size; (void)ws_size;

    const float* poses    = (const float*)d_in[0];
    const float* act      = (const float*)d_in[1];
    const float* Wt       = (const float*)d_in[2];
    const float* beta_v   = (const float*)d_in[3];
    const float* beta_a   = (const float*)d_in[4];
    const float* lambda_p = (const float*)d_in[5];
    float* out = (float*)d_out;

    float* ws  = (float*)d_ws;
    size_t off = 0;
    float* V   = ws + off; off += (size_t)NB * BKK * CWW * HH;
    float* R   = ws + off; off += (size_t)NB * BKK * CWW;
    float* ap  = ws + off; off += (size_t)NB * BKK * WW;
    float* mu  = ws + off; off += (size_t)NB * CWW * HH;
    float* sig = ws + off; off += (size_t)NB * CWW * HH;
    float* aA  = ws + off; off += (size_t)NB * CWW;

    // 1) activation patches
    {
        const int n = NB * BKK * WW;
        ap_kernel<<<(n + 255) / 256, 256, 0, stream>>>(act, ap);
    }
    // 2) votes via WMMA f32 16x16x4 (LDS-staged coalesced stores)
    votes_wmma_kernel<<<BKK * NSPLIT, 256, 0, stream>>>(poses, Wt, V);

    // 3) EM routing iterations
    for (int it = 0; it < ITER; ++it) {
        float* mu_dst = (it == ITER - 1) ? out : mu;
        float* a_dst  = (it == ITER - 1) ? (out + (size_t)NB * CWW * HH) : aA;
        em_stats_kernel<<<NB * CWW, 64, 0, stream>>>(
            V, R, ap, beta_v, beta_a, lambda_p, (it == 0) ? 1 : 0,
            mu_dst, sig, a_dst);
        if (it != ITER - 1) {
            em_rupdate_kernel<<<NB * BKK, 256, 0, stream>>>(V, mu, sig, aA, R);
        }
    }
}